// ResMambaBlock_6562710028934
// MI455X (gfx1250) — compile-verified
//
#include <hip/hip_runtime.h>
#include <hip/hip_bf16.h>

typedef __attribute__((ext_vector_type(2))) float v2f;
typedef __attribute__((ext_vector_type(8))) float v8f;

#define Bn   2
#define Cn   48
#define Ln   16384        // power of 2: / and % fold to shift/mask
#define DI   96
#define NST  16
#define RNK  3
#define XZC  192          // 2*DI
#define DBLC 36           // RNK + 2*NST = 35, padded to 36 for 8B alignment
#define NGRP 8

// ---------------------------------------------------------------------------
// GroupNorm stats: one block per (b, group); reduce mean / rstd over Cg*L.
// ---------------------------------------------------------------------------
__global__ void gn_stats_kernel(const float* __restrict__ x, float* __restrict__ stats) {
    __shared__ float ls[256];
    __shared__ float ls2[256];
    int bg = blockIdx.x;                 // 0 .. B*NGRP-1
    int g = bg % NGRP, b = bg / NGRP;
    const int cg = Cn / NGRP;            // 6
    const int n = cg * Ln;               // 98304
    int base = (b * Cn + g * cg) * Ln;
    float s = 0.f, s2 = 0.f;
    for (int i = threadIdx.x; i < n; i += blockDim.x) {
        float v = x[base + i];
        s += v; s2 += v * v;
    }
    ls[threadIdx.x] = s; ls2[threadIdx.x] = s2;
    __syncthreads();
    for (int stride = 128; stride > 0; stride >>= 1) {
        if (threadIdx.x < stride) {
            ls[threadIdx.x]  += ls[threadIdx.x + stride];
            ls2[threadIdx.x] += ls2[threadIdx.x + stride];
        }
        __syncthreads();
    }
    if (threadIdx.x == 0) {
        float m = ls[0] / (float)n;
        float var = ls2[0] / (float)n - m * m;
        stats[bg * 2 + 0] = m;
        stats[bg * 2 + 1] = rsqrtf(var + 1e-5f);
    }
}

__global__ void gn_apply_relu_kernel(const float* __restrict__ x, const float* __restrict__ stats,
                                     const float* __restrict__ w, const float* __restrict__ bb,
                                     float* __restrict__ out) {
    int idx = blockIdx.x * blockDim.x + threadIdx.x;
    const int total = Bn * Cn * Ln;
    if (idx >= total) return;
    int c = (idx / Ln) % Cn;
    int b = idx / (Cn * Ln);
    int bg = b * NGRP + c / (Cn / NGRP);
    float v = (x[idx] - stats[2 * bg]) * stats[2 * bg + 1] * w[c] + bb[c];
    out[idx] = fmaxf(v, 0.f);
}

// ---------------------------------------------------------------------------
// LayerNorm over C=48 per token. SRC_BCL=1 -> input in (B,C,L) layout
// (coalesced across lanes per c); else row-major (BL, C). Output (BL, C).
// ---------------------------------------------------------------------------
template <int SRC_BCL>
__global__ void layernorm_kernel(const float* __restrict__ src, float* __restrict__ dst,
                                 const float* __restrict__ w, const float* __restrict__ bp) {
    int row = blockIdx.x * blockDim.x + threadIdx.x;   // token id in [0, B*L)
    if (row >= Bn * Ln) return;
    int b = row / Ln, l = row % Ln;
    float s = 0.f, s2 = 0.f;
    #pragma unroll 4
    for (int c = 0; c < Cn; ++c) {
        float v = SRC_BCL ? src[(b * Cn + c) * Ln + l] : src[row * Cn + c];
        s += v; s2 += v * v;
    }
    float m = s / (float)Cn;
    float var = s2 / (float)Cn - m * m;
    float r = rsqrtf(var + 1e-5f);
    #pragma unroll 4
    for (int c = 0; c < Cn; ++c) {
        float v = SRC_BCL ? src[(b * Cn + c) * Ln + l] : src[row * Cn + c];
        dst[row * Cn + c] = (v - m) * r * w[c] + bp[c];
    }
}

// ---------------------------------------------------------------------------
// fp32 WMMA GEMM: Out tile = A[M,K] @ W^T  (W: (Nreal x Kw) row-major).
// One wave owns a 16-row M tile and NT n-tiles (n_base = blockIdx.y*NT*16):
// per k-step one unconditional b64 A-fragment load feeds NT
// v_wmma_f32_16x16x4_f32 ops. KT (k-steps) is compile time -> fully unrolled,
// all loads are saddr + 32-bit voffset + immediate, zero runtime guards
// unless NGUARD/KGUARD is set (then branchless clamp + cndmask).
// Fragment mapping per ISA 7.12.2:
//   A 16x4 : lanes 0-15 -> {K=k0,k0+1}; lanes 16-31 -> {K=k0+2,k0+3}
//   C/D    : acc[v] = C[m0 + 8*(lane>=16) + v][n + lane%16]
// MODE: 0 plain row-major store; 1 softplus(val+bias) row-major;
//       2 val += skip*res[(b,C,L)] row-major;
//       3 val += bias, store (B,C,L); 4 = mode 3 plus residual add
// ---------------------------------------------------------------------------
template <int NT, int KT, int MODE, bool KGUARD, bool NGUARD>
__global__ void wmma_gemm(const float* __restrict__ A, int lda, int Kreal,
                          const float* __restrict__ W, int Kw, int Nreal, int Nstore,
                          float* __restrict__ Out, int ldc,
                          const float* __restrict__ bias,
                          const float* __restrict__ res,
                          const float* __restrict__ skipv) {
    int lane = threadIdx.x & 31;
    int wv   = threadIdx.x >> 5;
    int half = lane >> 4;              // 0 | 1
    int lm   = lane & 15;
    int m0 = (blockIdx.x * (blockDim.x >> 5) + wv) * 16;
    int n_base = blockIdx.y * (NT * 16);

    v8f acc[NT];
    #pragma unroll
    for (int t = 0; t < NT; ++t) acc[t] = (v8f){};

    // 32-bit base offsets: loads become saddr + voffset + imm
    int aoff = (m0 + lm) * lda + half * 2;
    int woff[NT];
    #pragma unroll
    for (int t = 0; t < NT; ++t) {
        int ncol = n_base + t * 16 + lm;
        int nc = (NGUARD && ncol >= Nreal) ? (Nreal - 1) : ncol;   // clamp addr
        woff[t] = nc * Kw + half * 2;
    }

    #pragma unroll
    for (int kt = 0; kt < KT; ++kt) {
        v2f a = *(const v2f*)(A + aoff + kt * 4);
        v2f bf[NT];
        #pragma unroll
        for (int t = 0; t < NT; ++t) {
            if (KGUARD) {
                int k0 = kt * 4 + half * 2;          // dt case: Kw=3, scalar loads
                int ka = (k0     < Kreal) ? k0     : 0;
                int kb = (k0 + 1 < Kreal) ? k0 + 1 : 0;
                int ncol = n_base + t * 16 + lm;
                float w0 = W[ncol * Kw + ka];
                float w1 = W[ncol * Kw + kb];
                bf[t][0] = (k0     < Kreal) ? w0 : 0.f;
                bf[t][1] = (k0 + 1 < Kreal) ? w1 : 0.f;
            } else {
                v2f wld = *(const v2f*)(W + woff[t] + kt * 4);
                if (NGUARD) {
                    int ncol = n_base + t * 16 + lm;
                    bf[t][0] = (ncol < Nreal) ? wld[0] : 0.f;
                    bf[t][1] = (ncol < Nreal) ? wld[1] : 0.f;
                } else {
                    bf[t] = wld;
                }
            }
        }
        #pragma unroll
        for (int t = 0; t < NT; ++t)
            acc[t] = __builtin_amdgcn_wmma_f32_16x16x4_f32(false, a, false, bf[t],
                                                           (short)0, acc[t], false, false);
    }

    float sk = (MODE == 2) ? skipv[0] : 1.0f;
    #pragma unroll
    for (int t = 0; t < NT; ++t) {
        int ncol = n_base + t * 16 + lm;
        if (NGUARD && ncol >= Nstore) continue;
        #pragma unroll
        for (int v = 0; v < 8; ++v) {
            int m = m0 + half * 8 + v;
            int bi = m / Ln, l = m % Ln;            // shift/mask
            float val = acc[t][v];
            if (MODE == 1) {
                val += bias[ncol];
                val = (val > 20.f) ? val : __logf(1.f + __expf(val));   // softplus
            } else if (MODE == 2) {
                val += sk * res[(bi * Cn + ncol) * Ln + l];
            } else if (MODE == 3) {
                val += bias[ncol];
            } else if (MODE == 4) {
                val += bias[ncol] + res[(bi * Cn + ncol) * Ln + l];
            }
            if (MODE >= 3)
                Out[(bi * Cn + ncol) * Ln + l] = val;   // (B,C,L) store
            else
                Out[m * ldc + ncol] = val;
        }
    }
}

// ---------------------------------------------------------------------------
// Depthwise causal conv (k=4) over l, per channel d, + SiLU.  xi = xz[:, :DI].
// Branchless edge handling: clamp source index, zero the tap weight instead.
// ---------------------------------------------------------------------------
__global__ void conv_silu_kernel(const float* __restrict__ xz, const float* __restrict__ cw,
                                 const float* __restrict__ cb, float* __restrict__ u) {
    int idx = blockIdx.x * blockDim.x + threadIdx.x;
    const int total = Bn * Ln * DI;
    if (idx >= total) return;
    int d = idx % DI;
    int l = (idx / DI) % Ln;
    int b = idx / (DI * Ln);
    float acc = cb[d];
    #pragma unroll
    for (int k = 0; k < 4; ++k) {
        int ls = l - 3 + k;
        int lc = (ls >= 0) ? ls : 0;                       // clamp address
        float wv = (ls >= 0) ? cw[d * 4 + k] : 0.f;        // zero the tap
        acc += wv * xz[(b * Ln + lc) * XZC + d];
    }
    float sg = 1.f / (1.f + __expf(-acc));
    u[(b * Ln + l) * DI + d] = acc * sg;
}

// ---------------------------------------------------------------------------
// Chunked selective scan. One block per (b,d): 256 threads, each owns a
// contiguous chunk of 64 tokens and all 16 states in registers.
// Pass1: per-chunk (prod a, scanned b) -> LDS. Serial 16-lane combine over
// 256 chunks gives the exclusive prefix (h before chunk; h0 = 0). Pass2:
// replay chunk, fuse y = sum_n h*Cm + u*D, then * silu(z). dA/dBu are
// recomputed from delta/Bm/u (L2-resident) instead of materializing 400 MB.
// ---------------------------------------------------------------------------
__global__ void scan_kernel(const float* __restrict__ delta, const float* __restrict__ u,
                            const float* __restrict__ dbl, const float* __restrict__ xz,
                            const float* __restrict__ A_log, const float* __restrict__ Dp,
                            float* __restrict__ y) {
    __shared__ float sA[NST][256];
    __shared__ float sB[NST][256];
    int t = threadIdx.x;
    int bd = blockIdx.x;
    int d = bd % DI, b = bd / DI;
    const int CL = Ln / 256;             // 64 tokens per thread

    float Arow[NST];
    #pragma unroll
    for (int n = 0; n < NST; ++n) Arow[n] = -__expf(A_log[d * NST + n]);
    float Dd = Dp[d];

    float aP[NST], bS[NST];
    #pragma unroll
    for (int n = 0; n < NST; ++n) { aP[n] = 1.f; bS[n] = 0.f; }

    int rowBase = b * Ln + t * CL;
    for (int i = 0; i < CL; ++i) {
        int row = rowBase + i;
        __builtin_prefetch(delta + (row + 4) * DI + d, 0, 0);   // global_prefetch_b8
        __builtin_prefetch(u     + (row + 4) * DI + d, 0, 0);
        float dv = delta[row * DI + d];
        float uv = u[row * DI + d];
        const float* dblrow = dbl + row * DBLC;
        #pragma unroll
        for (int n = 0; n < NST; ++n) {
            float a  = __expf(dv * Arow[n]);
            float bv = dv * dblrow[RNK + n] * uv;        // delta * Bm * u
            aP[n] *= a;
            bS[n] = a * bS[n] + bv;
        }
    }
    #pragma unroll
    for (int n = 0; n < NST; ++n) { sA[n][t] = aP[n]; sB[n][t] = bS[n]; }
    __syncthreads();

    if (t < NST) {                        // exclusive cross-chunk scan, h0 = 0
        float ex = 0.f;
        for (int c = 0; c < 256; ++c) {
            float ta = sA[t][c], tb = sB[t][c];
            sB[t][c] = ex;                // h entering chunk c
            ex = ta * ex + tb;
        }
    }
    __syncthreads();

    float h[NST];
    #pragma unroll
    for (int n = 0; n < NST; ++n) h[n] = sB[n][t];

    for (int i = 0; i < CL; ++i) {
        int row = rowBase + i;
        float dv = delta[row * DI + d];
        float uv = u[row * DI + d];
        const float* dblrow = dbl + row * DBLC;
        float acc = uv * Dd;
        #pragma unroll
        for (int n = 0; n < NST; ++n) {
            float a  = __expf(dv * Arow[n]);
            float bv = dv * dblrow[RNK + n] * uv;
            h[n] = a * h[n] + bv;
            acc += h[n] * dblrow[RNK + NST + n];         // * Cm
        }
        float zv = xz[row * XZC + DI + d];
        float sil = zv / (1.f + __expf(-zv));            // silu(z)
        y[row * DI + d] = acc * sil;
    }
}

// ---------------------------------------------------------------------------
// Host orchestration
// ---------------------------------------------------------------------------
static void run_mamba_layer(const float* hin, float* hout, const float* residual,
                            const float* ln_w, const float* ln_b, const float* in_w,
                            const float* conv_w, const float* conv_b, const float* xp_w,
                            const float* dt_w, const float* dt_b, const float* A_log,
                            const float* Dp, const float* out_w, const float* proj_w,
                            const float* proj_b, const float* skipv,
                            float* xn, float* xz, float* u, float* dbl, float* delta,
                            float* yb, float* xm, float* xmn, hipStream_t stream) {
    const int BL = Bn * Ln;
    const int MB = BL / 64;              // blocks: 4 waves x 16-row tiles
    // layernorm of (B,C,L) input -> xn (BL,48)
    layernorm_kernel<1><<<(BL + 255) / 256, 256, 0, stream>>>(hin, xn, ln_w, ln_b);
    // in_proj: (BL,48)@(192x48)^T -> xz ; grid.y=2 n-halves of 96
    wmma_gemm<6, 12, 0, false, false><<<dim3(MB, 2), 128, 0, stream>>>(
        xn, Cn, Cn, in_w, Cn, XZC, XZC, xz, XZC, nullptr, nullptr, nullptr);
    // depthwise conv + silu -> u (BL,96)
    conv_silu_kernel<<<(BL * DI + 255) / 256, 256, 0, stream>>>(xz, conv_w, conv_b, u);
    // x_proj: (BL,96)@(35x96)^T -> dbl (ld 36; cols 35..47 zero via B zero-fill)
    wmma_gemm<3, 24, 0, false, true><<<MB, 128, 0, stream>>>(
        u, DI, DI, xp_w, DI, 35, DBLC, dbl, DBLC, nullptr, nullptr, nullptr);
    // dt_proj + softplus: (BL,3)@(96x3)^T, K padded to 4 (B row k=3 zeroed)
    wmma_gemm<6, 1, 1, true, false><<<MB, 128, 0, stream>>>(
        dbl, DBLC, RNK, dt_w, RNK, DI, DI, delta, DI, dt_b, nullptr, nullptr);
    // selective scan -> y (BL,96), fused with Cm-contraction, u*D and silu(z)
    scan_kernel<<<Bn * DI, 256, 0, stream>>>(delta, u, dbl, xz, A_log, Dp, yb);
    // out_proj + skip*xf: (BL,96)@(48x96)^T -> xm (BL,48)
    wmma_gemm<3, 24, 2, false, false><<<MB, 128, 0, stream>>>(
        yb, DI, DI, out_w, DI, Cn, Cn, xm, Cn, nullptr, hin, skipv);
    // second layernorm (same weights per reference), row-major input
    layernorm_kernel<0><<<(BL + 255) / 256, 256, 0, stream>>>(xm, xmn, ln_w, ln_b);
    // final proj + bias (+ optional global residual), stored back to (B,C,L)
    if (residual)
        wmma_gemm<3, 12, 4, false, false><<<MB, 128, 0, stream>>>(
            xmn, Cn, Cn, proj_w, Cn, Cn, Cn, hout, Cn, proj_b, residual, nullptr);
    else
        wmma_gemm<3, 12, 3, false, false><<<MB, 128, 0, stream>>>(
            xmn, Cn, Cn, proj_w, Cn, Cn, Cn, hout, Cn, proj_b, nullptr, nullptr);
}

extern "C" void kernel_launch(void* const* d_in, const int* in_sizes, int n_in,
                              void* d_out, int out_size, void* d_ws, size_t ws_size,
                              hipStream_t stream) {
    const float* x     = (const float*)d_in[0];
    const float* gn1_w = (const float*)d_in[1];
    const float* gn1_b = (const float*)d_in[2];
    const float* gn2_w = (const float*)d_in[3];
    const float* gn2_b = (const float*)d_in[4];
    const float* p1[14], *p2[14];
    for (int i = 0; i < 14; ++i) { p1[i] = (const float*)d_in[5 + i]; p2[i] = (const float*)d_in[19 + i]; }
    // p[]: 0 ln_w 1 ln_b 2 in_w 3 conv_w 4 conv_b 5 xp_w 6 dt_w 7 dt_b 8 A_log 9 Dp 10 out_w 11 proj_w 12 proj_b 13 skip

    const int BL = Bn * Ln;
    const int BCL = Bn * Cn * Ln;

    float* ws = (float*)d_ws;
    size_t off = 0;
    float* stats = ws + off; off += 64;
    float* H0    = ws + off; off += BCL;          // gn output / layer input
    float* H1    = ws + off; off += BCL;          // layer-1 output
    float* xn    = ws + off; off += (size_t)BL * Cn;
    float* xz    = ws + off; off += (size_t)BL * XZC;
    float* u     = ws + off; off += (size_t)BL * DI;
    float* dbl   = ws + off; off += (size_t)BL * DBLC;
    float* delta = ws + off; off += (size_t)BL * DI;
    float* yb    = ws + off; off += (size_t)BL * DI;
    float* xm    = ws + off; off += (size_t)BL * Cn;
    float* xmn   = ws + off; off += (size_t)BL * Cn;   // ~99 MB total

    // h = relu(groupnorm1(x))
    gn_stats_kernel<<<Bn * NGRP, 256, 0, stream>>>(x, stats);
    gn_apply_relu_kernel<<<(BCL + 255) / 256, 256, 0, stream>>>(x, stats, gn1_w, gn1_b, H0);
    // layer 1
    run_mamba_layer(H0, H1, nullptr,
                    p1[0], p1[1], p1[2], p1[3], p1[4], p1[5], p1[6], p1[7], p1[8], p1[9],
                    p1[10], p1[11], p1[12], p1[13],
                    xn, xz, u, dbl, delta, yb, xm, xmn, stream);
    // h = relu(groupnorm2(h))
    gn_stats_kernel<<<Bn * NGRP, 256, 0, stream>>>(H1, stats);
    gn_apply_relu_kernel<<<(BCL + 255) / 256, 256, 0, stream>>>(H1, stats, gn2_w, gn2_b, H0);
    // layer 2 (+ global identity residual fused into final projection store)
    run_mamba_layer(H0, (float*)d_out, x,
                    p2[0], p2[1], p2[2], p2[3], p2[4], p2[5], p2[6], p2[7], p2[8], p2[9],
                    p2[10], p2[11], p2[12], p2[13],
                    xn, xz, u, dbl, delta, yb, xm, xmn, stream);
}